// GNNThicknessPredictor_75299366633991
// MI455X (gfx1250) — compile-verified
//
#include <hip/hip_runtime.h>
#include <math.h>

// ---------------------------------------------------------------------------
// GATv2 GNN forward for MI455X (gfx1250, wave32, WMMA f32 16x16x4)
//   N=50000 nodes, E=800000 edges (+N self loops), HID=128, H=4 heads, C=32
// ---------------------------------------------------------------------------

#define HID   128
#define NHEAD 4
#define OUTD  8

typedef float v2f __attribute__((ext_vector_type(2)));
typedef float v8f __attribute__((ext_vector_type(8)));

// ---- helpers --------------------------------------------------------------

// monotone float <-> uint mapping so atomicMax(uint) == float max
__device__ __forceinline__ unsigned f_ord(float f) {
  unsigned u = __float_as_uint(f);
  return (u & 0x80000000u) ? ~u : (u | 0x80000000u);
}
__device__ __forceinline__ float f_unord(unsigned u) {
  return (u & 0x80000000u) ? __uint_as_float(u & 0x7fffffffu)
                           : __uint_as_float(~u);
}
__device__ __forceinline__ float gelu_exact(float x) {
  return 0.5f * x * (1.0f + erff(x * 0.70710678118654752f));
}

// ---- generic fill ---------------------------------------------------------

__global__ void fill_kernel(float* __restrict__ p, float v, long long n) {
  long long i = (long long)blockIdx.x * blockDim.x + threadIdx.x;
  if (i < n) p[i] = v;
}

// ---- self-loop attr: deg + sum -------------------------------------------

__global__ void deg_kernel(const int* __restrict__ ei, const float* __restrict__ ea,
                           float* __restrict__ deg, float* __restrict__ loopsum, int E_) {
  int e = blockIdx.x * blockDim.x + threadIdx.x;
  if (e >= E_) return;
  int d = ei[E_ + e];                 // dst row of edge_index (2,E)
  atomicAdd(&deg[d], 1.0f);
  atomicAdd(&loopsum[d], ea[e]);
}

__global__ void loopattr_kernel(const float* __restrict__ deg, float* __restrict__ loopsum,
                                int N_) {
  int n = blockIdx.x * blockDim.x + threadIdx.x;
  if (n >= N_) return;
  loopsum[n] = loopsum[n] / fmaxf(deg[n], 1.0f);   // finalize in place
}

// ---- f32 WMMA GEMM:  C[M,Nc] = A[M,K] @ B[K,Nc] + bias  -------------------
// One wave per 16x16 output tile; K stepped by 4 via V_WMMA_F32_16X16X4_F32.

__global__ void gemm_wmma_f32(const float* __restrict__ A, const float* __restrict__ B,
                              const float* __restrict__ bias, float* __restrict__ C,
                              int M, int K, int Nc) {
  int wave = (int)((blockIdx.x * (long long)blockDim.x + threadIdx.x) >> 5);
  int lane = threadIdx.x & 31;
  int ntiles = Nc >> 4;
  int mt = wave / ntiles;
  int nt = wave - mt * ntiles;
  if (mt * 16 >= M) return;          // wave-uniform guard (EXEC stays all-ones)

  int half = lane >> 4;              // 0: K pair {0,1}, 1: K pair {2,3}
  int lid  = lane & 15;
  int arow = mt * 16 + lid;          // A row handled by this lane
  int bcol = nt * 16 + lid;          // B col handled by this lane

  v8f acc = {};
  for (int k = 0; k < K; k += 4) {
    int ka = k + half * 2;
    v2f a, b;
    a[0] = A[(long long)arow * K + ka];
    a[1] = A[(long long)arow * K + ka + 1];
    b[0] = B[(long long)ka * Nc + bcol];
    b[1] = B[(long long)(ka + 1) * Nc + bcol];
    acc = __builtin_amdgcn_wmma_f32_16x16x4_f32(false, a, false, b,
                                                (short)0, acc, false, false);
  }
  float bv = bias ? bias[bcol] : 0.0f;
#pragma unroll
  for (int r = 0; r < 8; ++r) {
    int orow = mt * 16 + r + half * 8;     // C/D: VGPR r -> row r (+8 for lanes 16-31)
    C[(long long)orow * Nc + bcol] = acc[r] + bv;
  }
}

// ---- node post-process: (raw [/denom] [+bias]) -> LN -> GELU [-> +res] ----
// One wave per node; lane l owns channels 4l..4l+3 (head = l>>3).

__global__ void node_post_kernel(const float* __restrict__ raw,
                                 const float* __restrict__ denom,
                                 const float* __restrict__ bias,
                                 const float* __restrict__ g,
                                 const float* __restrict__ b,
                                 float* __restrict__ h,
                                 int N_, int use_res) {
  int wave = (int)((blockIdx.x * (long long)blockDim.x + threadIdx.x) >> 5);
  int lane = threadIdx.x & 31;
  if (wave >= N_) return;

  const float4 v4 = *(const float4*)(raw + (long long)wave * HID + lane * 4);
  float vx = v4.x, vy = v4.y, vz = v4.z, vw = v4.w;
  if (denom) {
    float inv = 1.0f / (denom[wave * NHEAD + (lane >> 3)] + 1e-16f);
    vx *= inv; vy *= inv; vz *= inv; vw *= inv;
  }
  if (bias) {
    const float4 bb = *(const float4*)(bias + lane * 4);
    vx += bb.x; vy += bb.y; vz += bb.z; vw += bb.w;
  }
  // mean over 128
  float s = vx + vy + vz + vw;
#pragma unroll
  for (int m = 16; m >= 1; m >>= 1) s += __shfl_xor(s, m, 32);
  float mean = s * (1.0f / 128.0f);
  float dx = vx - mean, dy = vy - mean, dz = vz - mean, dw = vw - mean;
  float q = dx * dx + dy * dy + dz * dz + dw * dw;
#pragma unroll
  for (int m = 16; m >= 1; m >>= 1) q += __shfl_xor(q, m, 32);
  float rstd = rsqrtf(q * (1.0f / 128.0f) + 1e-5f);

  const float4 gg = *(const float4*)(g + lane * 4);
  const float4 be = *(const float4*)(b + lane * 4);
  float ox = gelu_exact(dx * rstd * gg.x + be.x);
  float oy = gelu_exact(dy * rstd * gg.y + be.y);
  float oz = gelu_exact(dz * rstd * gg.z + be.z);
  float ow = gelu_exact(dw * rstd * gg.w + be.w);

  float* hp = h + (long long)wave * HID + lane * 4;
  if (use_res) {
    float4 hv = *(const float4*)hp;
    ox += hv.x; oy += hv.y; oz += hv.z; ow += hv.w;
  }
  float4 o; o.x = ox; o.y = oy; o.z = oz; o.w = ow;
  *(float4*)hp = o;
}

// ---- edge pass 1: raw attention logits + per-dst max ----------------------
// One wave per edge (incl. self loops). lane l: channels 4l..4l+3, head l>>3.

__global__ void edge_alpha_kernel(const int* __restrict__ ei,
                                  const float* __restrict__ ea,
                                  const float* __restrict__ loopat,
                                  const float* __restrict__ xl,
                                  const float* __restrict__ xr,
                                  const float* __restrict__ we,
                                  const float* __restrict__ att,
                                  float* __restrict__ alpha,
                                  unsigned* __restrict__ amax,
                                  int E_, int N_) {
  int wave = (int)((blockIdx.x * (long long)blockDim.x + threadIdx.x) >> 5);
  int lane = threadIdx.x & 31;
  if (wave >= E_ + N_) return;
  int s, d; float eav;
  if (wave < E_) { s = ei[wave]; d = ei[E_ + wave]; eav = ea[wave]; }
  else           { s = d = wave - E_; eav = loopat[s]; }

  const float4 a4 = *(const float4*)(xl + (long long)s * HID + lane * 4);
  const float4 r4 = *(const float4*)(xr + (long long)d * HID + lane * 4);
  const float4 w4 = *(const float4*)(we + lane * 4);
  const float4 t4 = *(const float4*)(att + lane * 4);   // att flat (H*C)=128

  float mx = a4.x + r4.x + eav * w4.x;
  float my = a4.y + r4.y + eav * w4.y;
  float mz = a4.z + r4.z + eav * w4.z;
  float mw = a4.w + r4.w + eav * w4.w;
  mx = mx > 0.0f ? mx : 0.2f * mx;
  my = my > 0.0f ? my : 0.2f * my;
  mz = mz > 0.0f ? mz : 0.2f * mz;
  mw = mw > 0.0f ? mw : 0.2f * mw;
  float p = mx * t4.x + my * t4.y + mz * t4.z + mw * t4.w;
  // reduce within each 8-lane head group
  p += __shfl_xor(p, 1, 32);
  p += __shfl_xor(p, 2, 32);
  p += __shfl_xor(p, 4, 32);
  if ((lane & 7) == 0) {
    int head = lane >> 3;
    alpha[(long long)wave * NHEAD + head] = p;
    atomicMax(&amax[(long long)d * NHEAD + head], f_ord(p));
  }
}

// ---- edge pass 2: exp(alpha - max) + denominator --------------------------

__global__ void edge_exp_kernel(const int* __restrict__ ei,
                                float* __restrict__ alpha,
                                const unsigned* __restrict__ amax,
                                float* __restrict__ denom,
                                int E_, int N_) {
  long long t = (long long)blockIdx.x * blockDim.x + threadIdx.x;
  long long tot = (long long)(E_ + N_) * NHEAD;
  if (t >= tot) return;
  int e = (int)(t >> 2), hh = (int)(t & 3);
  int d = (e < E_) ? ei[E_ + e] : (e - E_);
  float a = expf(alpha[t] - f_unord(amax[(long long)d * NHEAD + hh]));
  alpha[t] = a;
  atomicAdd(&denom[(long long)d * NHEAD + hh], a);
}

// ---- edge pass 3: unnormalized weighted scatter ---------------------------
// out_raw[dst] += xl[src] * exp_alpha  (divide by denom happens in node_post)

__global__ void edge_scatter_kernel(const int* __restrict__ ei,
                                    const float* __restrict__ alpha,
                                    const float* __restrict__ xl,
                                    float* __restrict__ raw,
                                    int E_, int N_) {
  int wave = (int)((blockIdx.x * (long long)blockDim.x + threadIdx.x) >> 5);
  int lane = threadIdx.x & 31;
  if (wave >= E_ + N_) return;
  int s, d;
  if (wave < E_) { s = ei[wave]; d = ei[E_ + wave]; }
  else           { s = d = wave - E_; }
  float w = alpha[(long long)wave * NHEAD + (lane >> 3)];
  const float4 v = *(const float4*)(xl + (long long)s * HID + lane * 4);
  float* op = raw + (long long)d * HID + lane * 4;
  atomicAdd(op + 0, v.x * w);
  atomicAdd(op + 1, v.y * w);
  atomicAdd(op + 2, v.z * w);
  atomicAdd(op + 3, v.w * w);
}

// ---- output heads: softplus(h@tw+tb) and h@ew+eb --------------------------

__global__ void heads_kernel(const float* __restrict__ h,
                             const float* __restrict__ tw, const float* __restrict__ tb,
                             const float* __restrict__ ew, const float* __restrict__ eb,
                             float* __restrict__ out, int N_) {
  long long t = (long long)blockIdx.x * blockDim.x + threadIdx.x;
  if (t >= (long long)N_ * 16) return;
  int n = (int)(t >> 4), o = (int)(t & 15);
  const float* hp = h + (long long)n * HID;
  if (o < OUTD) {
    float s = tb[o];
    for (int k = 0; k < HID; ++k) s += hp[k] * tw[k * OUTD + o];
    out[(long long)n * OUTD + o] = fmaxf(s, 0.0f) + log1pf(expf(-fabsf(s)));  // softplus
  } else {
    int oo = o - OUTD;
    float s = eb[oo];
    for (int k = 0; k < HID; ++k) s += hp[k] * ew[k * OUTD + oo];
    out[(long long)N_ * OUTD + (long long)n * OUTD + oo] = s;
  }
}

// ---------------------------------------------------------------------------

extern "C" void kernel_launch(void* const* d_in, const int* in_sizes, int n_in,
                              void* d_out, int out_size, void* d_ws, size_t ws_size,
                              hipStream_t stream) {
  const float* x  = (const float*)d_in[0];
  const int*   ei = (const int*)d_in[1];
  const float* ea = (const float*)d_in[2];
  const int N_ = in_sizes[0] / 32;     // 50000
  const int E_ = in_sizes[2];          // 800000

  // params flattened in jax pytree order (sorted dict keys, lists in order)
  int pi = 3;
  const float* exist_b = (const float*)d_in[pi++];
  const float* exist_w = (const float*)d_in[pi++];
  const float* in_b    = (const float*)d_in[pi++];
  const float* in_ln_b = (const float*)d_in[pi++];
  const float* in_ln_g = (const float*)d_in[pi++];
  const float* in_w    = (const float*)d_in[pi++];
  const float *att[3], *bias[3], *bl[3], *br[3], *ln_b[3], *ln_g[3], *we[3], *wl[3], *wr[3];
  for (int l = 0; l < 3; ++l) {
    att[l]  = (const float*)d_in[pi++];
    bias[l] = (const float*)d_in[pi++];
    bl[l]   = (const float*)d_in[pi++];
    br[l]   = (const float*)d_in[pi++];
    ln_b[l] = (const float*)d_in[pi++];
    ln_g[l] = (const float*)d_in[pi++];
    we[l]   = (const float*)d_in[pi++];
    wl[l]   = (const float*)d_in[pi++];
    wr[l]   = (const float*)d_in[pi++];
  }
  const float* thick_b = (const float*)d_in[pi++];
  const float* thick_w = (const float*)d_in[pi++];

  // workspace layout (floats)
  float* ws = (float*)d_ws;
  long long NH  = (long long)N_ * HID;
  float*    h     = ws;                       // N*128
  float*    xl    = h    + NH;                // N*128
  float*    xr    = xl   + NH;                // N*128
  float*    raw   = xr   + NH;                // N*128
  float*    alpha = raw  + NH;                // (E+N)*4
  unsigned* amax  = (unsigned*)(alpha + (long long)(E_ + N_) * NHEAD);  // N*4
  float*    denom = (float*)amax + (long long)N_ * NHEAD;               // N*4
  float*    deg   = denom + (long long)N_ * NHEAD;                      // N
  float*    loopa = deg + N_;                                           // N
  (void)ws_size; (void)n_in; (void)out_size;

  const int T = 256;
  int EP = E_ + N_;                               // edges incl. self loops
  dim3 blkFillNH((unsigned)((NH + T - 1) / T));
  dim3 blkNode((unsigned)((N_ + 7) / 8));         // 8 waves/block, wave per node
  dim3 blkEdgeW((unsigned)((EP + 7) / 8));        // wave per edge
  dim3 blkEdgeT((unsigned)(((long long)EP * NHEAD + T - 1) / T));
  dim3 blkE((unsigned)((E_ + T - 1) / T));
  dim3 blkN((unsigned)((N_ + T - 1) / T));
  int mtiles = (N_ + 15) / 16;
  long long gw = (long long)mtiles * (HID / 16) * 32;   // gemm threads
  dim3 blkGemm((unsigned)((gw + T - 1) / T));

  // --- self loop attributes -------------------------------------------------
  fill_kernel<<<dim3((unsigned)((2LL * N_ + T - 1) / T)), T, 0, stream>>>(deg, 0.0f, 2LL * N_);
  deg_kernel<<<blkE, T, 0, stream>>>(ei, ea, deg, loopa, E_);
  loopattr_kernel<<<blkN, T, 0, stream>>>(deg, loopa, N_);

  // --- input projection + LN + GELU ----------------------------------------
  gemm_wmma_f32<<<dim3((unsigned)(((long long)mtiles * (HID/16) * 32 + T - 1) / T)), T, 0, stream>>>(
      x, in_w, in_b, raw, N_, 32, HID);
  node_post_kernel<<<blkNode, T, 0, stream>>>(raw, nullptr, nullptr, in_ln_g, in_ln_b, h, N_, 0);

  // --- GATv2 layers ---------------------------------------------------------
  for (int l = 0; l < 3; ++l) {
    gemm_wmma_f32<<<blkGemm, T, 0, stream>>>(h, wl[l], bl[l], xl, N_, HID, HID);
    gemm_wmma_f32<<<blkGemm, T, 0, stream>>>(h, wr[l], br[l], xr, N_, HID, HID);
    // zero amax (ordered-uint 0 == minimum), denom, raw
    fill_kernel<<<dim3((unsigned)((2LL * N_ * NHEAD + T - 1) / T)), T, 0, stream>>>(
        (float*)amax, 0.0f, 2LL * N_ * NHEAD);
    fill_kernel<<<blkFillNH, T, 0, stream>>>(raw, 0.0f, NH);

    edge_alpha_kernel<<<blkEdgeW, T, 0, stream>>>(ei, ea, loopa, xl, xr, we[l], att[l],
                                                  alpha, amax, E_, N_);
    edge_exp_kernel<<<blkEdgeT, T, 0, stream>>>(ei, alpha, amax, denom, E_, N_);
    edge_scatter_kernel<<<blkEdgeW, T, 0, stream>>>(ei, alpha, xl, raw, E_, N_);
    node_post_kernel<<<blkNode, T, 0, stream>>>(raw, denom, bias[l], ln_g[l], ln_b[l], h, N_, 1);
  }

  // --- output heads ---------------------------------------------------------
  heads_kernel<<<dim3((unsigned)(((long long)N_ * 16 + T - 1) / T)), T, 0, stream>>>(
      h, thick_w, thick_b, exist_w, exist_b, (float*)d_out, N_);
}